// WaveletTransform_44092134260934
// MI455X (gfx1250) — compile-verified
//
#include <hip/hip_runtime.h>

#ifndef __has_builtin
#define __has_builtin(x) 0
#endif

// ---------------------------------------------------------------------------
// db4 wavelet features for MI455X (gfx1250).
// One workgroup (256 threads = 8 wave32) per signal of 16384 f32.
// Signal is DMA'd into LDS with the CDNA5 Tensor Data Mover (tensor_load_to_lds,
// TENSORcnt), cascade runs entirely in LDS (ping-pong buffers), detail bands are
// consumed on the fly (never stored), features reduced with wave32 shuffles.
// ---------------------------------------------------------------------------

namespace {

constexpr int NT  = 256;       // threads per block
constexpr int NW  = NT / 32;   // wave32 waves per block
constexpr int CH  = 8;         // outputs per thread per chunk
constexpr int WIN = 2 * CH + 8; // 24-float register window (incl. 2 extra for prev output)

constexpr int T0 = 16384;                 // input length
constexpr int PQ_OFF = T0;                // Q buffer offset in floats
constexpr int QLEN   = 8200;              // >= 8195, padded
constexpr int RED_OFF = T0 + QLEN;        // reduction scratch
constexpr int SMEM_FLOATS = RED_OFF + 64; // 98,592 bytes total

// pywt db4 decomposition filters (dec_lo / dec_hi)
constexpr float DLO[8] = {
  -0.010597401784997278f,  0.032883011666982945f,  0.030841381835986965f,
  -0.18703481171888114f,  -0.02798376941698385f,   0.6308807679295904f,
   0.7148465705525415f,    0.23037781330885523f };
constexpr float DHI[8] = {
  -0.23037781330885523f,   0.7148465705525415f,   -0.6308807679295904f,
  -0.02798376941698385f,   0.18703481171888114f,   0.030841381835986965f,
  -0.032883011666982945f, -0.010597401784997278f };

struct Stats { float s, s2, sa, mx, zc; };

__device__ __forceinline__ void accum(Stats& st, float v) {
  st.s  += v;
  st.s2  = fmaf(v, v, st.s2);
  float av = fabsf(v);
  st.sa += av;
  st.mx  = fmaxf(st.mx, av);
}

__device__ __forceinline__ float sgnf(float v) {
  return (float)((v > 0.0f) - (v < 0.0f)); // matches jnp.sign (-1/0/+1)
}

__device__ __forceinline__ float wredsum(float v) {
#pragma unroll
  for (int o = 16; o > 0; o >>= 1) v += __shfl_down(v, o, 32);
  return v;
}
__device__ __forceinline__ float wredmax(float v) {
#pragma unroll
  for (int o = 16; o > 0; o >>= 1) v = fmaxf(v, __shfl_down(v, o, 32));
  return v;
}

// Block-level reduction of 5 partial stats -> 6 features written by thread 0.
__device__ void stats_finalize(Stats st, float invL, float* red, float* out6) {
  st.s  = wredsum(st.s);
  st.s2 = wredsum(st.s2);
  st.sa = wredsum(st.sa);
  st.mx = wredmax(st.mx);
  st.zc = wredsum(st.zc);
  const int lane = threadIdx.x & 31;
  const int wid  = threadIdx.x >> 5;
  if (lane == 0) {
    red[wid * 5 + 0] = st.s;
    red[wid * 5 + 1] = st.s2;
    red[wid * 5 + 2] = st.sa;
    red[wid * 5 + 3] = st.mx;
    red[wid * 5 + 4] = st.zc;
  }
  __syncthreads();  // also serves as the inter-level LDS hazard barrier
  if (threadIdx.x == 0) {
    float S = 0.f, S2 = 0.f, SA = 0.f, MX = 0.f, ZC = 0.f;
#pragma unroll
    for (int w = 0; w < NW; ++w) {
      S  += red[w * 5 + 0];
      S2 += red[w * 5 + 1];
      SA += red[w * 5 + 2];
      MX  = fmaxf(MX, red[w * 5 + 3]);
      ZC += red[w * 5 + 4];
    }
    const float mean = S * invL;
    const float ms   = S2 * invL;
    float var = ms - mean * mean;
    var = var > 0.f ? var : 0.f;
    out6[0] = SA * invL;   // mean_abs
    out6[1] = sqrtf(var);  // population std
    out6[2] = sqrtf(ms);   // rms
    out6[3] = MX;          // max |c|
    out6[4] = S2;          // energy
    out6[5] = ZC * invL;   // zero-crossing rate
  }
  __syncthreads();  // protect `red` before next level's leaders rewrite it
}

// One DWT level. pywt symmetric mode: o[i] = sum_j h[j] * x[sym(2i+1-j)],
// sym(p) = -1-p for p<0, 2L-1-p for p>=L.  LOUT = (LIN+7)/2.
// Detail (hi) features accumulated on the fly; cA stored to outA unless LAST,
// in which case cA features are also accumulated (cA4 + cD4).
template <int LIN, bool LAST>
__device__ void dwt_level(const float* __restrict__ in, float* __restrict__ outA,
                          float* red, float* gout6D, float* gout6A) {
  constexpr int LOUT = (LIN + 7) >> 1;
  Stats sd = {0.f, 0.f, 0.f, 0.f, 0.f};
  Stats sa = {0.f, 0.f, 0.f, 0.f, 0.f};

  for (int b0 = threadIdx.x * CH; b0 < LOUT; b0 += NT * CH) {
    float w[WIN];
    const int base = 2 * b0 - 8;  // window covers in[2*b0-8 .. 2*b0+15]
    if (base >= 0 && base + WIN <= LIN) {
#pragma unroll
      for (int k = 0; k < WIN; ++k) w[k] = in[base + k];
    } else {
#pragma unroll
      for (int k = 0; k < WIN; ++k) {
        int p = base + k;
        p = (p < 0) ? (-1 - p) : p;
        p = (p >= LIN) ? (2 * LIN - 1 - p) : p;
        w[k] = in[p];
      }
    }

    // previous output (i = b0-1) for chunk-boundary zero-crossing pair
    float pd = 0.f, pa = 0.f;
#pragma unroll
    for (int j = 0; j < 8; ++j) pd = fmaf(DHI[j], w[7 - j], pd);
    if (LAST) {
#pragma unroll
      for (int j = 0; j < 8; ++j) pa = fmaf(DLO[j], w[7 - j], pa);
    }

#pragma unroll
    for (int u = 0; u < CH; ++u) {
      const int i = b0 + u;
      if (i < LOUT) {
        float lo = 0.f, hi = 0.f;
#pragma unroll
        for (int j = 0; j < 8; ++j) {
          const float xv = w[2 * u + 9 - j];
          lo = fmaf(DLO[j], xv, lo);
          hi = fmaf(DHI[j], xv, hi);
        }
        if (!LAST) outA[i] = lo;
        accum(sd, hi);
        if (i > 0) sd.zc += (sgnf(pd) != sgnf(hi)) ? 1.f : 0.f;
        pd = hi;
        if (LAST) {
          accum(sa, lo);
          if (i > 0) sa.zc += (sgnf(pa) != sgnf(lo)) ? 1.f : 0.f;
          pa = lo;
        }
      }
    }
  }

  const float invL = 1.0f / (float)LOUT;
  stats_finalize(sd, invL, red, gout6D);
  if (LAST) stats_finalize(sa, invL, red, gout6A);
}

}  // namespace

// --- CDNA5 Tensor Data Mover path ------------------------------------------
// This toolchain (clang-23 + therock-10.0 headers) declares the 6-arg form:
//   (uint32x4 g0, int32x8 g1, int32x4 g2, int32x4 g3, int32x8 g4, i32 cpol)
#if __has_builtin(__builtin_amdgcn_tensor_load_to_lds) && \
    __has_builtin(__builtin_amdgcn_s_wait_tensorcnt)
#define USE_TDM 1
typedef unsigned int v4u __attribute__((ext_vector_type(4)));
typedef int          v8i __attribute__((ext_vector_type(8)));
typedef int          v4i __attribute__((ext_vector_type(4)));
#endif

extern "C" __global__ void __launch_bounds__(NT)
wavelet30(const float* __restrict__ x, float* __restrict__ out) {
  extern __shared__ float smem[];
  float* P   = smem;             // 16384 floats (input / cA2)
  float* Q   = smem + PQ_OFF;    // 8200 floats  (cA1 / cA3)
  float* red = smem + RED_OFF;   // 64 floats    (reduction scratch)

  const float* gx  = x + (size_t)blockIdx.x * T0;
  float*      gout = out + (size_t)blockIdx.x * 30;

#ifdef USE_TDM
  // One TDM descriptor: contiguous 16384 x f32 (64 KB) tile, global -> LDS.
  if (threadIdx.x == 0) {
    const unsigned long long ga = (unsigned long long)(uintptr_t)gx;
    const unsigned lds_off = (unsigned)(uintptr_t)(void*)P;  // low 32 bits = LDS offset
    // D# group0: count=1 | lds_addr | global_addr[56:0] | type=2 (bits 127:126)
    v4u g0 = { 1u,
               lds_off,
               (unsigned)ga,
               ((unsigned)(ga >> 32) & 0x01FFFFFFu) | 0x80000000u };
    // D# group1: data_size=4B; tensor_dim0=16384; tensor_dim1=1; tile_dim0=16384;
    //            tile_dim1=1; tensor_dim0_stride=16384; no pad/iterate/multicast.
    v8i g1 = { (int)(2u << 16),        // [17:16] data_size = 2 (4 bytes)
               (int)(16384u << 16),    // [63:48] tensor_dim0 lo16
               (int)(1u << 16),        // [79:64] dim0 hi16=0, [95:80] tensor_dim1 lo16=1
               (int)(16384u << 16),    // [111:96] dim1 hi16=0, [127:112] tile_dim0
               1,                      // [143:128] tile_dim1 = 1, [159:144] tile_dim2 = 0
               16384,                  // [191:160] tensor_dim0_stride lo32
               0, 0 };
    v4i gz4 = { 0, 0, 0, 0 };          // groups 2/3: dims 2..4 unused
    v8i gz8 = { 0, 0, 0, 0, 0, 0, 0, 0 };
    __builtin_amdgcn_tensor_load_to_lds(g0, g1, gz4, gz4, gz8, 0);
    __builtin_amdgcn_s_wait_tensorcnt(0);
  }
  __syncthreads();
#else
  {
    const float4* g4 = (const float4*)gx;
    float4*       p4 = (float4*)P;
    for (int k = threadIdx.x; k < T0 / 4; k += NT) p4[k] = g4[k];
  }
  __syncthreads();
#endif

  // Cascade: lengths are compile-time constants -> fully unrolled filters.
  // coeff order in output: [cA4 | cD4 | cD3 | cD2 | cD1] * 6 features.
  dwt_level<16384, false>(P, Q, red, gout + 24, nullptr);  // cD1 -> slot 4
  dwt_level< 8195, false>(Q, P, red, gout + 18, nullptr);  // cD2 -> slot 3
  dwt_level< 4101, false>(P, Q, red, gout + 12, nullptr);  // cD3 -> slot 2
  dwt_level< 2054, true >(Q, nullptr, red, gout + 6, gout + 0); // cD4, cA4
}

extern "C" void kernel_launch(void* const* d_in, const int* in_sizes, int n_in,
                              void* d_out, int out_size, void* d_ws, size_t ws_size,
                              hipStream_t stream) {
  (void)n_in; (void)d_ws; (void)ws_size; (void)out_size;
  const float* x = (const float*)d_in[0];
  float* out = (float*)d_out;
  const int nsig = in_sizes[0] / T0;  // 64*64 = 4096 signals
  const size_t shmem = (size_t)SMEM_FLOATS * sizeof(float);  // ~98.6 KB
  (void)hipFuncSetAttribute((const void*)wavelet30,
                            hipFuncAttributeMaxDynamicSharedMemorySize, (int)shmem);
  wavelet30<<<nsig, NT, shmem, stream>>>(x, out);
}